// GCN_77034533421326
// MI455X (gfx1250) — compile-verified
//
#include <hip/hip_runtime.h>

typedef __attribute__((ext_vector_type(2))) float v2f;
typedef __attribute__((ext_vector_type(8))) float v8f;

#define D_IN  128
#define UNITS 128

// ---------------- degree / normalization ----------------

__global__ void deg_init_kernel(float* __restrict__ deg, int N) {
    int i = blockIdx.x * blockDim.x + threadIdx.x;
    if (i < N) deg[i] = 1.0f;   // self-loop fill weight
}

__global__ void deg_acc_kernel(const int* __restrict__ row,
                               const float* __restrict__ w,
                               float* __restrict__ deg, int E) {
    int e = blockIdx.x * blockDim.x + threadIdx.x;
    if (e < E) unsafeAtomicAdd(&deg[row[e]], w[e]);
}

__global__ void deg_inv_sqrt_kernel(float* __restrict__ deg, int N) {
    int i = blockIdx.x * blockDim.x + threadIdx.x;
    if (i < N) {
        float d = deg[i];
        float r = (d > 0.0f) ? rsqrtf(d) : 0.0f;
        // one Newton-Raphson step -> fp32-accurate d^-1/2 (r==0 stays 0)
        r = r * (1.5f - 0.5f * d * r * r);
        deg[i] = r;
    }
}

// ---------------- dense transform h = x @ W via V_WMMA_F32_16X16X4_F32 ----------------
// One wave computes a 16(nodes) x 16(units) tile; 8 waves/block cover all 128 units
// for a 16-node tile. x tile staged in LDS (row pitch 132 -> conflict-free b64 reads).

__global__ __launch_bounds__(256) void gemm_wmma_kernel(const float* __restrict__ x,
                                                        const float* __restrict__ W,
                                                        float* __restrict__ h, int N) {
    __shared__ float xs[16 * 132];
    const int tid  = threadIdx.x;
    const int base = blockIdx.x * 16;

    // cooperative load: 16 x 128 floats = 512 float4, 2 per thread (clamped rows)
    #pragma unroll
    for (int i = 0; i < 2; ++i) {
        int idx  = tid + i * 256;     // float4 index
        int r    = idx >> 5;          // 32 float4 per row
        int c4   = idx & 31;
        int grow = base + r; if (grow > N - 1) grow = N - 1;
        const float4 v = *(const float4*)(x + (size_t)grow * D_IN + c4 * 4);
        float* dst = xs + r * 132 + c4 * 4;
        dst[0] = v.x; dst[1] = v.y; dst[2] = v.z; dst[3] = v.w;
    }
    __syncthreads();

    const int wave = tid >> 5;
    const int lane = tid & 31;
    const int m    = lane & 15;   // row within tile (A) / column (B,C,D)
    const int hi   = lane >> 4;   // half-wave selector
    const int nb   = wave * 16;   // unit-column base for this wave

    v8f acc = {};
    const float* xr = xs + m * 132 + 2 * hi;                    // A: x[m][k + 2*hi ..]
    const float* wp = W + (size_t)(2 * hi) * UNITS + nb + m;    // B: W[k + 2*hi][nb+m]

    #pragma unroll
    for (int k = 0; k < D_IN; k += 4) {
        v2f a;                                    // A 16x4 f32 layout (ISA 7.12.2)
        a.x = xr[k];
        a.y = xr[k + 1];
        v2f b;                                    // B 4x16 f32 layout (row across lanes)
        b.x = wp[(size_t)k * UNITS];
        b.y = wp[(size_t)(k + 1) * UNITS];
        acc = __builtin_amdgcn_wmma_f32_16x16x4_f32(
            /*neg_a=*/false, a, /*neg_b=*/false, b,
            /*c_mod=*/(short)0, acc, /*reuse_a=*/false, /*reuse_b=*/false);
    }

    // C/D layout: VGPR g -> row (g + 8*hi), col m
    #pragma unroll
    for (int g = 0; g < 8; ++g) {
        int rown = base + g + 8 * hi;
        if (rown < N) h[(size_t)rown * UNITS + nb + m] = acc[g];
    }
}

// ---------------- out = bias + selfloop + edge scatter ----------------

__global__ void out_init_kernel(const float* __restrict__ h,
                                const float* __restrict__ dis,
                                const float* __restrict__ bias,
                                float* __restrict__ out, int total) {
    int i = blockIdx.x * blockDim.x + threadIdx.x;   // over N*128
    if (i < total) {
        int n = i >> 7;
        int c = i & 127;
        float s = dis[n];
        out[i] = bias[c] + s * s * h[i];             // self-loop: w=1, norm = dis[n]^2
    }
}

__global__ __launch_bounds__(256) void spmm_scatter_kernel(const int* __restrict__ row,
                                                           const int* __restrict__ col,
                                                           const float* __restrict__ w,
                                                           const float* __restrict__ dis,
                                                           const float* __restrict__ h,
                                                           float* __restrict__ out, int E) {
    int e = blockIdx.x * 8 + (threadIdx.x >> 5);     // one wave32 per edge
    if (e >= E) return;
    int lane = threadIdx.x & 31;                     // 4 channels per lane
    int r = row[e];
    int c = col[e];
    float coeff = dis[r] * w[e] * dis[c];
    const float4 hv = *(const float4*)(h + (size_t)c * UNITS + lane * 4);
    float* o = out + (size_t)r * UNITS + lane * 4;
    unsafeAtomicAdd(o + 0, coeff * hv.x);
    unsafeAtomicAdd(o + 1, coeff * hv.y);
    unsafeAtomicAdd(o + 2, coeff * hv.z);
    unsafeAtomicAdd(o + 3, coeff * hv.w);
}

// ---------------- launch ----------------

extern "C" void kernel_launch(void* const* d_in, const int* in_sizes, int n_in,
                              void* d_out, int out_size, void* d_ws, size_t ws_size,
                              hipStream_t stream) {
    const float* x    = (const float*)d_in[0];
    const int*   ei   = (const int*)d_in[1];     // [2, E]: row = ei[0:E], col = ei[E:2E]
    const float* ew   = (const float*)d_in[2];
    const float* W    = (const float*)d_in[3];
    const float* bias = (const float*)d_in[4];
    float* out = (float*)d_out;

    const int N = in_sizes[0] / D_IN;
    const int E = in_sizes[1] / 2;

    float* h   = (float*)d_ws;                   // N*128 floats (~51.2 MB)
    float* deg = h + (size_t)N * UNITS;          // N floats

    const int* row = ei;
    const int* col = ei + E;

    deg_init_kernel<<<(N + 255) / 256, 256, 0, stream>>>(deg, N);
    deg_acc_kernel<<<(E + 255) / 256, 256, 0, stream>>>(row, ew, deg, E);
    deg_inv_sqrt_kernel<<<(N + 255) / 256, 256, 0, stream>>>(deg, N);

    gemm_wmma_kernel<<<(N + 15) / 16, 256, 0, stream>>>(x, W, h, N);

    int total = N * UNITS;
    out_init_kernel<<<(total + 255) / 256, 256, 0, stream>>>(h, deg, bias, out, total);
    spmm_scatter_kernel<<<(E + 7) / 8, 256, 0, stream>>>(row, col, ew, deg, h, out, E);
}